// GCN_8297876816695
// MI455X (gfx1250) — compile-verified
//
#include <hip/hip_runtime.h>
#include <hip/hip_bf16.h>
#include <stdint.h>

// Fixed by the reference problem: feature width 128 floats = 32 x float4 = one wave32 row.
#define FD 128
#define EPB 256   // edges staged per block (1 per thread; each wave consumes its own 32)

// ---------------------------------------------------------------------------
// x[row] = features[row] / max(||features[row]||_2, 1e-12)
// One wave per row: lane loads float4 (global_load_b128), wave32 shfl reduction.
__global__ void gcn_normalize_kernel(const float* __restrict__ f,
                                     float* __restrict__ x, int N) {
  int warp = threadIdx.x >> 5;
  int lane = threadIdx.x & 31;
  int row  = blockIdx.x * (blockDim.x >> 5) + warp;
  if (row >= N) return;
  float4 v = ((const float4*)(f + (size_t)row * FD))[lane];
  float ss = v.x * v.x + v.y * v.y + v.z * v.z + v.w * v.w;
#pragma unroll
  for (int off = 16; off > 0; off >>= 1) ss += __shfl_xor(ss, off, 32);
  float scale = 1.0f / fmaxf(sqrtf(ss), 1e-12f);
  float4 o = {v.x * scale, v.y * scale, v.z * scale, v.w * scale};
  ((float4*)(x + (size_t)row * FD))[lane] = o;
}

// deg[src] += 1.0f for non-self-loop edges (matches segment_sum of float mask).
__global__ void gcn_degree_kernel(const int* __restrict__ src,
                                  const int* __restrict__ dst,
                                  float* __restrict__ deg, int E) {
  int e = blockIdx.x * blockDim.x + threadIdx.x;
  if (e >= E) return;
  int s = src[e], d = dst[e];
  if (s != d) atomicAdd(&deg[s], 1.0f);
}

// dis[i] = deg>0 ? rsqrt(deg) : 0
__global__ void gcn_dis_kernel(const float* __restrict__ deg,
                               float* __restrict__ dis, int N) {
  int i = blockIdx.x * blockDim.x + threadIdx.x;
  if (i >= N) return;
  float dg = deg[i];
  dis[i] = dg > 0.0f ? rsqrtf(dg) : 0.0f;
}

// out[dst] += dis[src]*dis[dst] * in[src]   (self-loops masked via w==0)
// Stage: each thread async-loads its edge's (src,dst) into LDS via CDNA5
// global_load_async_to_lds_b32, drains its own ASYNCcnt, then computes the
// per-edge weight once into LDS. Each wave stages exactly the 32 edges it
// consumes -> no cross-wave dependency, no __syncthreads needed.
// Consume: one wave per edge row (128 floats = 32 lanes x float4), 2x unrolled
// so two row gathers are in flight ahead of the atomic bursts.
__global__ void gcn_conv_kernel(const float* __restrict__ in,
                                float* __restrict__ out,
                                const int* __restrict__ src,
                                const int* __restrict__ dst,
                                const float* __restrict__ dis, int E) {
  __shared__ int   s_src[EPB];
  __shared__ int   s_dst[EPB];
  __shared__ float s_w[EPB];
  int t    = threadIdx.x;
  int warp = t >> 5;
  int lane = t & 31;
  int base = blockIdx.x * EPB;

  int eg  = base + t;
  int egc = eg < E ? eg : (E - 1);   // clamp; clamped lanes get w=0 below
  // Low 32 bits of a generic pointer to __shared__ == wave-relative LDS offset
  // (LDS aperture base has zero low bits; async op adds LDS_BASE in hardware).
  unsigned lds_s = (unsigned)(size_t)&s_src[t];
  unsigned lds_d = (unsigned)(size_t)&s_dst[t];
  unsigned long long ga_s = (unsigned long long)(size_t)(src + egc);
  unsigned long long ga_d = (unsigned long long)(size_t)(dst + egc);
  asm volatile("global_load_async_to_lds_b32 %0, %2, off\n\t"
               "global_load_async_to_lds_b32 %1, %3, off"
               :: "v"(lds_s), "v"(lds_d), "v"(ga_s), "v"(ga_d)
               : "memory");
  asm volatile("s_wait_asynccnt 0" ::: "memory");
  __builtin_amdgcn_wave_barrier();

  // Per-edge norm computed once (not once per lane): dis gathers drop 32x.
  {
    int sv = s_src[t];
    int dv = s_dst[t];
    float wv = 0.0f;
    if (eg < E && sv != dv) wv = dis[sv] * dis[dv];  // 0 if either deg==0
    s_w[t] = wv;
  }
  __builtin_amdgcn_wave_barrier();  // same-wave LDS producer->consumer ordering

  int nloc = E - base;
  if (nloc > EPB) nloc = EPB;
  int wbeg = warp * 32;
  int wend = wbeg + 32;
  if (wend > nloc) wend = nloc;

  int i = wbeg;
  for (; i + 1 < wend; i += 2) {
    float w0 = s_w[i];         // wave-uniform LDS broadcast
    float w1 = s_w[i + 1];
    int   s0 = s_src[i],     d0 = s_dst[i];
    int   s1 = s_src[i + 1], d1 = s_dst[i + 1];
    float4 v0, v1;
    if (w0 != 0.0f) v0 = ((const float4*)(in + (size_t)s0 * FD))[lane];
    if (w1 != 0.0f) v1 = ((const float4*)(in + (size_t)s1 * FD))[lane];
    if (w0 != 0.0f) {
      float* o = out + (size_t)d0 * FD + lane * 4;
      atomicAdd(o + 0, w0 * v0.x);
      atomicAdd(o + 1, w0 * v0.y);
      atomicAdd(o + 2, w0 * v0.z);
      atomicAdd(o + 3, w0 * v0.w);
    }
    if (w1 != 0.0f) {
      float* o = out + (size_t)d1 * FD + lane * 4;
      atomicAdd(o + 0, w1 * v1.x);
      atomicAdd(o + 1, w1 * v1.y);
      atomicAdd(o + 2, w1 * v1.z);
      atomicAdd(o + 3, w1 * v1.w);
    }
  }
  if (i < wend) {
    float w0 = s_w[i];
    if (w0 != 0.0f) {
      int s0 = s_src[i], d0 = s_dst[i];
      float4 v0 = ((const float4*)(in + (size_t)s0 * FD))[lane];
      float* o = out + (size_t)d0 * FD + lane * 4;
      atomicAdd(o + 0, w0 * v0.x);
      atomicAdd(o + 1, w0 * v0.y);
      atomicAdd(o + 2, w0 * v0.z);
      atomicAdd(o + 3, w0 * v0.w);
    }
  }
}

// d_out[0:ND) = x + h   (conv2 then atomically accumulates h1 on top)
// d_out[ND:ND+NUD) = preference (passthrough second tuple element)
__global__ void gcn_combine_kernel(const float* __restrict__ x,
                                   const float* __restrict__ h,
                                   const float* __restrict__ pref,
                                   float* __restrict__ out,
                                   long long nd4, long long tot4) {
  long long i = (long long)blockIdx.x * blockDim.x + threadIdx.x;
  if (i >= tot4) return;
  float4* o = (float4*)out;
  if (i < nd4) {
    float4 a = ((const float4*)x)[i];
    float4 b = ((const float4*)h)[i];
    float4 r = {a.x + b.x, a.y + b.y, a.z + b.z, a.w + b.w};
    o[i] = r;
  } else {
    o[i] = ((const float4*)pref)[i - nd4];
  }
}

extern "C" void kernel_launch(void* const* d_in, const int* in_sizes, int n_in,
                              void* d_out, int out_size, void* d_ws, size_t ws_size,
                              hipStream_t stream) {
  const float* features   = (const float*)d_in[0];
  const float* preference = (const float*)d_in[1];
  const int*   edge       = (const int*)d_in[2];
  // d_in[3] (edge_index_drop) is unused by the reference.

  const int N  = in_sizes[0] / FD;
  const int NU = in_sizes[1] / FD;
  const int E  = in_sizes[2] / 2;
  const int* src = edge;       // edge_index[0]
  const int* dst = edge + E;   // edge_index[1]

  // Workspace layout: x[N*D] | h[N*D] | deg[N] | dis[N]  (~52 MB)
  float* x   = (float*)d_ws;
  float* h   = x + (size_t)N * FD;
  float* deg = h + (size_t)N * FD;
  float* dis = deg + N;

  hipMemsetAsync(h,   0, (size_t)N * FD * sizeof(float), stream);
  hipMemsetAsync(deg, 0, (size_t)N * sizeof(float), stream);

  gcn_normalize_kernel<<<(N + 7) / 8, 256, 0, stream>>>(features, x, N);
  gcn_degree_kernel<<<(E + 255) / 256, 256, 0, stream>>>(src, dst, deg, E);
  gcn_dis_kernel<<<(N + 255) / 256, 256, 0, stream>>>(deg, dis, N);

  // h = conv(x)
  gcn_conv_kernel<<<(E + EPB - 1) / EPB, 256, 0, stream>>>(x, h, src, dst, dis, E);

  // d_out = [x + h, preference]
  long long nd4  = (long long)N * FD / 4;
  long long tot4 = nd4 + (long long)NU * FD / 4;
  gcn_combine_kernel<<<(int)((tot4 + 255) / 256), 256, 0, stream>>>(
      x, h, preference, (float*)d_out, nd4, tot4);

  // d_out[0:ND) += conv(h)  ==  x + h + h1
  gcn_conv_kernel<<<(E + EPB - 1) / EPB, 256, 0, stream>>>(h, (float*)d_out, src, dst, dis, E);
}